// GATNet_76158360092713
// MI455X (gfx1250) — compile-verified
//
#include <hip/hip_runtime.h>
#include <hip/hip_bf16.h>
#include <stdint.h>

// ---------------------------------------------------------------------------
// GAT 2-layer network for gfx1250 (MI455X).
// GEMMs use V_WMMA_F32_16X16X4_F32 (full f32 precision, wave32 layouts).
// Edge softmax/aggregation uses native f32 atomic add + u32 atomic max.
// ---------------------------------------------------------------------------

typedef __attribute__((ext_vector_type(2))) float v2f;
typedef __attribute__((ext_vector_type(8))) float v8f;

#define DIMC 128          // feature width everywhere (IN_DIM = HEADS*HID = OUT_DIM = 128)
#define HEADS1 8
#define HID1 16
#define NEG_SLOPE 0.2f
#define MAP_NEG_INF 0x007FFFFFu   // fmap(-inf)

// order-preserving float <-> u32 mapping for atomic max on floats
__device__ __forceinline__ unsigned fmap(float f) {
  unsigned u = __float_as_uint(f);
  return (u >> 31) ? ~u : (u | 0x80000000u);
}
__device__ __forceinline__ float funmap(unsigned u) {
  return __uint_as_float((u >> 31) ? (u & 0x7FFFFFFFu) : ~u);
}
__device__ __forceinline__ float lrelu(float l) {
  return l > 0.0f ? l : NEG_SLOPE * l;
}

// ---------------------------------------------------------------------------
// C[M,128] = A[M,128] @ B[128,128]  via V_WMMA_F32_16X16X4_F32.
// Block = 256 threads = 8 waves; block b owns M-tile b, wave w owns N-tile w.
// A 16x4 layout: lanes 0-15 -> M=lane, hi=lane/16 selects K pair {2hi, 2hi+1}.
// B 4x16 layout: lanes 0-15 -> N=lane, v0/v1 hold K rows {2hi, 2hi+1}.
// C/D: VGPR j, lane L -> row = j + 8*(L/16), col = L%16.
// ---------------------------------------------------------------------------
__global__ __launch_bounds__(256) void gemm128_wmma_f32(
    const float* __restrict__ A, const float* __restrict__ B,
    float* __restrict__ C, int M) {
  const int wave = threadIdx.x >> 5;
  const int lane = threadIdx.x & 31;
  const int lo = lane & 15;
  const int hi = lane >> 4;
  const int m0 = blockIdx.x << 4;
  const int n0 = wave << 4;
  if (m0 >= M) return;

  const float* __restrict__ arow = A + (size_t)(m0 + lo) * DIMC;
  v8f acc = {};
#pragma unroll
  for (int k0 = 0; k0 < DIMC; k0 += 4) {
    const int ka = k0 + 2 * hi;
    v2f a, b;
    a.x = arow[ka];
    a.y = arow[ka + 1];
    b.x = B[(size_t)ka * DIMC + n0 + lo];
    b.y = B[(size_t)(ka + 1) * DIMC + n0 + lo];
    acc = __builtin_amdgcn_wmma_f32_16x16x4_f32(
        /*neg_a=*/false, a, /*neg_b=*/false, b,
        /*c_mod=*/(short)0, acc, /*reuse_a=*/false, /*reuse_b=*/false);
  }
  float* __restrict__ crow = C + (size_t)(m0 + 8 * hi) * DIMC + n0 + lo;
#pragma unroll
  for (int j = 0; j < 8; ++j) crow[(size_t)j * DIMC] = acc[j];
}

// ---------------------------------------------------------------------------
// Layer-1 per-node prep: attention logits (8 heads, 16-dot each), zero the
// aggregation buffer / denominators, init running max.  Block = node, 128 thr.
// ---------------------------------------------------------------------------
__global__ __launch_bounds__(128) void l1_prep(
    const float* __restrict__ h1, const float* __restrict__ a_src,
    const float* __restrict__ a_dst, float* __restrict__ al_s,
    float* __restrict__ al_d, unsigned* __restrict__ emax,
    float* __restrict__ den, float* __restrict__ agg) {
  const int n = blockIdx.x;
  const int t = threadIdx.x;
  agg[(size_t)n * DIMC + t] = 0.0f;
  if (t < HEADS1) {
    const float* __restrict__ hrow = h1 + (size_t)n * DIMC + t * HID1;
    float s = 0.0f, d = 0.0f;
#pragma unroll
    for (int c = 0; c < HID1; ++c) {
      const float hv = hrow[c];
      s += hv * a_src[t * HID1 + c];
      d += hv * a_dst[t * HID1 + c];
    }
    al_s[(size_t)n * HEADS1 + t] = s;
    al_d[(size_t)n * HEADS1 + t] = d;
    den[(size_t)n * HEADS1 + t] = 0.0f;
    emax[(size_t)n * HEADS1 + t] = MAP_NEG_INF;
  }
}

// ---- layer-1 edge pass A: running max of leaky-relu logits per (dst, head) -
__global__ __launch_bounds__(256) void l1_edge_max(
    const long long* __restrict__ ei, long long E, long long Etot,
    const float* __restrict__ al_s, const float* __restrict__ al_d,
    unsigned* __restrict__ emax) {
  const long long gid = (long long)blockIdx.x * blockDim.x + threadIdx.x;
  if (gid >= Etot * HEADS1) return;
  const long long e = gid >> 3;
  const int head = (int)(gid & 7);
  long long s, d;
  if (e < E) { s = ei[e]; d = ei[E + e]; } else { s = d = e - E; }
  const float l = lrelu(al_s[s * HEADS1 + head] + al_d[d * HEADS1 + head]);
  atomicMax(&emax[d * HEADS1 + head], fmap(l));
}

// ---- layer-1 edge pass B: unnormalized scatter  agg[dst] += e * h1[src] ----
__global__ __launch_bounds__(128) void l1_edge_scatter(
    const long long* __restrict__ ei, long long E,
    const float* __restrict__ h1, const float* __restrict__ al_s,
    const float* __restrict__ al_d, const unsigned* __restrict__ emax,
    float* __restrict__ den, float* __restrict__ agg) {
  const long long e = blockIdx.x;
  const int t = threadIdx.x;
  const int head = t >> 4;
  long long s, d;
  if (e < E) { s = ei[e]; d = ei[E + e]; } else { s = d = e - E; }
  float l = lrelu(al_s[s * HEADS1 + head] + al_d[d * HEADS1 + head]);
  const float w = __expf(l - funmap(emax[d * HEADS1 + head]));
  atomicAdd(&agg[d * DIMC + t], w * h1[s * DIMC + t]);
  if ((t & 15) == 0) atomicAdd(&den[d * HEADS1 + head], w);
}

// ---- layer-1 finish: normalize + bias + relu (in place); prep layer 2 ------
__global__ __launch_bounds__(128) void l1_finish(
    float* __restrict__ agg /* becomes hact */, const float* __restrict__ den,
    const float* __restrict__ b1, float* __restrict__ out,
    unsigned* __restrict__ emax2, float* __restrict__ den2) {
  const int n = blockIdx.x;
  const int t = threadIdx.x;
  const float v = agg[(size_t)n * DIMC + t] / den[(size_t)n * HEADS1 + (t >> 4)] + b1[t];
  agg[(size_t)n * DIMC + t] = fmaxf(v, 0.0f);
  out[(size_t)n * DIMC + t] = 0.0f;
  if (t == 0) {
    den2[n] = 0.0f;
    emax2[n] = MAP_NEG_INF;
  }
}

// ---- layer-2 per-node logits (single head, 128-dot) ------------------------
__global__ __launch_bounds__(256) void l2_logits(
    const float* __restrict__ h2, const float* __restrict__ a_src,
    const float* __restrict__ a_dst, float* __restrict__ al_s,
    float* __restrict__ al_d, int N) {
  const int n = blockIdx.x * blockDim.x + threadIdx.x;
  if (n >= N) return;
  const float* __restrict__ hrow = h2 + (size_t)n * DIMC;
  float s = 0.0f, d = 0.0f;
#pragma unroll 8
  for (int c = 0; c < DIMC; ++c) {
    const float hv = hrow[c];
    s += hv * a_src[c];
    d += hv * a_dst[c];
  }
  al_s[n] = s;
  al_d[n] = d;
}

// ---- layer-2 edge pass A ---------------------------------------------------
__global__ __launch_bounds__(256) void l2_edge_max(
    const long long* __restrict__ ei, long long E, long long Etot,
    const float* __restrict__ al_s, const float* __restrict__ al_d,
    unsigned* __restrict__ emax) {
  const long long e = (long long)blockIdx.x * blockDim.x + threadIdx.x;
  if (e >= Etot) return;
  long long s, d;
  if (e < E) { s = ei[e]; d = ei[E + e]; } else { s = d = e - E; }
  atomicMax(&emax[d], fmap(lrelu(al_s[s] + al_d[d])));
}

// ---- layer-2 edge pass B: scatter into d_out -------------------------------
__global__ __launch_bounds__(128) void l2_edge_scatter(
    const long long* __restrict__ ei, long long E,
    const float* __restrict__ h2, const float* __restrict__ al_s,
    const float* __restrict__ al_d, const unsigned* __restrict__ emax,
    float* __restrict__ den, float* __restrict__ out) {
  const long long e = blockIdx.x;
  const int t = threadIdx.x;
  long long s, d;
  if (e < E) { s = ei[e]; d = ei[E + e]; } else { s = d = e - E; }
  const float w = __expf(lrelu(al_s[s] + al_d[d]) - funmap(emax[d]));
  atomicAdd(&out[d * DIMC + t], w * h2[s * DIMC + t]);
  if (t == 0) atomicAdd(&den[d], w);
}

// ---- layer-2 finish: normalize + bias --------------------------------------
__global__ __launch_bounds__(128) void l2_finish(
    float* __restrict__ out, const float* __restrict__ den,
    const float* __restrict__ b2) {
  const int n = blockIdx.x;
  const int t = threadIdx.x;
  out[(size_t)n * DIMC + t] = out[(size_t)n * DIMC + t] / den[n] + b2[t];
}

// ---------------------------------------------------------------------------
extern "C" void kernel_launch(void* const* d_in, const int* in_sizes, int n_in,
                              void* d_out, int out_size, void* d_ws, size_t ws_size,
                              hipStream_t stream) {
  const float* x       = (const float*)d_in[0];
  const long long* ei  = (const long long*)d_in[1];  // int64 [2, E]
  const float* W1      = (const float*)d_in[2];
  const float* a_src1  = (const float*)d_in[3];
  const float* a_dst1  = (const float*)d_in[4];
  const float* b1      = (const float*)d_in[5];
  const float* W2      = (const float*)d_in[6];
  const float* a_src2  = (const float*)d_in[7];
  const float* a_dst2  = (const float*)d_in[8];
  const float* b2      = (const float*)d_in[9];
  float* out = (float*)d_out;

  const int N = in_sizes[0] / DIMC;
  const long long E = (long long)in_sizes[1] / 2;
  const long long Etot = E + N;

  // workspace carve-up (floats)
  float* ws = (float*)d_ws;
  float*    h1    = ws;                                   // N*128
  float*    hact  = h1    + (size_t)N * DIMC;             // N*128 (agg1 -> hact)
  float*    h2    = hact  + (size_t)N * DIMC;             // N*128
  float*    al_s1 = h2    + (size_t)N * DIMC;             // N*8
  float*    al_d1 = al_s1 + (size_t)N * HEADS1;           // N*8
  unsigned* emax1 = (unsigned*)(al_d1 + (size_t)N * HEADS1); // N*8
  float*    den1  = (float*)(emax1 + (size_t)N * HEADS1); // N*8
  float*    al_s2 = den1  + (size_t)N * HEADS1;           // N
  float*    al_d2 = al_s2 + N;                            // N
  unsigned* emax2 = (unsigned*)(al_d2 + N);               // N
  float*    den2  = (float*)(emax2 + N);                  // N

  const int mtiles = (N + 15) / 16;

  // ---------------- layer 1 ----------------
  gemm128_wmma_f32<<<mtiles, 256, 0, stream>>>(x, W1, h1, N);
  l1_prep<<<N, 128, 0, stream>>>(h1, a_src1, a_dst1, al_s1, al_d1, emax1, den1, hact);
  {
    long long work = Etot * HEADS1;
    int blocks = (int)((work + 255) / 256);
    l1_edge_max<<<blocks, 256, 0, stream>>>(ei, E, Etot, al_s1, al_d1, emax1);
  }
  l1_edge_scatter<<<(unsigned)Etot, 128, 0, stream>>>(ei, E, h1, al_s1, al_d1,
                                                      emax1, den1, hact);
  l1_finish<<<N, 128, 0, stream>>>(hact, den1, b1, out, emax2, den2);

  // ---------------- layer 2 ----------------
  gemm128_wmma_f32<<<mtiles, 256, 0, stream>>>(hact, W2, h2, N);
  l2_logits<<<(N + 255) / 256, 256, 0, stream>>>(h2, a_src2, a_dst2, al_s2, al_d2, N);
  {
    int blocks = (int)((Etot + 255) / 256);
    l2_edge_max<<<blocks, 256, 0, stream>>>(ei, E, Etot, al_s2, al_d2, emax2);
  }
  l2_edge_scatter<<<(unsigned)Etot, 128, 0, stream>>>(ei, E, h2, al_s2, al_d2,
                                                      emax2, den2, out);
  l2_finish<<<N, 128, 0, stream>>>(out, den2, b2);
}